// DataStar_51204600103635
// MI455X (gfx1250) — compile-verified
//
#include <hip/hip_runtime.h>
#include <hip/hip_bf16.h>
#include <math.h>

// ---------------------------------------------------------------------------
// SMPL forward for MI455X (gfx1250, wave32).
// Big GEMMs run on V_WMMA_F32_16X16X4_F32, one wave = 16x32 output tile
// (2 accumulators sharing one A fragment). All inner-loop loads are
// unconditional (row-clamped / zero-padded) so there is no EXEC juggling.
// Everything else (J-regressor, kinematic chain, skinning) is VALU and is
// negligible next to the ~160MB memory-bound traffic (@23.3 TB/s ~7us).
// ---------------------------------------------------------------------------

#define N_VERTS 6890
#define N_JOINTS 24
#define BATCH 512
#define MROWS (N_VERTS * 3)      // 20670
#define KSHAPE 300
#define KPOSE 93

typedef float v2f __attribute__((ext_vector_type(2)));
typedef float v8f __attribute__((ext_vector_type(8)));

// One wave computes a 16x32 f32 tile of  C = A(MxK) * B^T  where B is stored
// as (N x LDB) row-major (B_gemm[k][n] = Bsrc[n*LDB + k]).
// Requirements:
//  - A row index is CLAMPED to M-1 (garbage rows are simply not stored).
//  - If KD % 4 != 0, Bsrc must be zero-padded up to a multiple of 4 within
//    LDB (tail contributions become A_garbage * 0 = 0).
// Per the ISA layouts, A-frag and B-frag share per-lane k = kb + 2*(lane>>4).
template <int KD, int LDA, int LDB>
__device__ inline void wmma_tile2_f32(const float* __restrict__ A,
                                      const float* __restrict__ Bsrc,
                                      int M, int m0, int n0, int lane,
                                      v8f& acc0, v8f& acc1) {
    const int row  = lane & 15;
    const int koff = (lane >> 4) * 2;
    int arow = m0 + row;
    if (arow > M - 1) arow = M - 1;                    // clamp: loads always legal
    const float* __restrict__ Ap  = A    + (size_t)arow * LDA + koff;
    const float* __restrict__ Bp0 = Bsrc + (size_t)(n0 + row) * LDB + koff;
    const float* __restrict__ Bp1 = Bsrc + (size_t)(n0 + 16 + row) * LDB + koff;

    constexpr int KMAIN = KD & ~3;
    acc0 = (v8f){}; acc1 = (v8f){};
#pragma unroll 4
    for (int kb = 0; kb < KMAIN; kb += 4) {
        v2f a, b0, b1;
        a.x  = Ap[kb];  a.y  = Ap[kb + 1];
        b0.x = Bp0[kb]; b0.y = Bp0[kb + 1];
        b1.x = Bp1[kb]; b1.y = Bp1[kb + 1];
        if (kb + 64 < KMAIN) __builtin_prefetch(Ap + kb + 64, 0, 3);
        acc0 = __builtin_amdgcn_wmma_f32_16x16x4_f32(
            false, a, false, b0, (short)0, acc0, false, false);
        acc1 = __builtin_amdgcn_wmma_f32_16x16x4_f32(
            false, a, false, b1, (short)0, acc1, false, false);
    }
    if constexpr (KMAIN < KD) {
        // Tail: B is zero beyond KD (caller-padded), A address clamped to KD-1.
        const int k0  = KMAIN + koff;
        const int kc0 = (k0     < KD) ? k0     : (KD - 1);
        const int kc1 = (k0 + 1 < KD) ? k0 + 1 : (KD - 1);
        const float* __restrict__ Ab = A + (size_t)arow * LDA;
        v2f a, b0, b1;
        a.x  = Ab[kc0]; a.y = Ab[kc1];
        b0.x = Bp0[KMAIN]; b0.y = Bp0[KMAIN + 1];
        b1.x = Bp1[KMAIN]; b1.y = Bp1[KMAIN + 1];
        acc0 = __builtin_amdgcn_wmma_f32_16x16x4_f32(
            false, a, false, b0, (short)0, acc0, false, false);
        acc1 = __builtin_amdgcn_wmma_f32_16x16x4_f32(
            false, a, false, b1, (short)0, acc1, false, false);
    }
}

// --- Kernel 1: v_shaped[b, m] = (shapedirs_flat @ betas^T)[m, b] + v_template[m]
__global__ void vshaped_kernel(const float* __restrict__ SD,
                               const float* __restrict__ betas,
                               const float* __restrict__ vt,
                               float* __restrict__ out_vs,
                               int Mtiles, int totalTiles) {
    const int wid  = threadIdx.x >> 5;
    const int lane = threadIdx.x & 31;
    const int tile = blockIdx.x * (blockDim.x >> 5) + wid;
    if (tile >= totalTiles) return;            // wave-uniform: EXEC stays full
    const int mt = tile % Mtiles;
    const int nt = tile / Mtiles;
    const int m0 = mt * 16, n0 = nt * 32;

    v8f acc0, acc1;
    wmma_tile2_f32<KSHAPE, KSHAPE, KSHAPE>(SD, betas, MROWS, m0, n0, lane,
                                           acc0, acc1);

    const int col  = lane & 15;
    const int hi   = lane >> 4;
    const int mb   = m0 + hi * 8;              // 8 consecutive rows per lane
    const float* __restrict__ vtp = vt + mb;
    float* __restrict__ o0 = out_vs + (size_t)(n0 + col) * MROWS + mb;
    float* __restrict__ o1 = out_vs + (size_t)(n0 + 16 + col) * MROWS + mb;
    if (m0 + 16 <= MROWS) {                    // full tile: unguarded stores
#pragma unroll
        for (int r = 0; r < 8; ++r) {
            const float t = vtp[r];
            o0[r] = acc0[r] + t;
            o1[r] = acc1[r] + t;
        }
    } else {
#pragma unroll
        for (int r = 0; r < 8; ++r) {
            if (mb + r < MROWS) {
                const float t = vtp[r];
                o0[r] = acc0[r] + t;
                o1[r] = acc1[r] + t;
            }
        }
    }
}

// --- Kernel 2: J[b, j, :] = sum_n J_regressor[j, n] * v_shaped[b, n, :]
__global__ void jreg_kernel(const float* __restrict__ vs,
                            const float* __restrict__ JR,
                            float* __restrict__ Jout) {
    __shared__ float r0[256], r1[256], r2[256];
    const int b = blockIdx.x / N_JOINTS;
    const int j = blockIdx.x % N_JOINTS;
    const int tid = threadIdx.x;
    const float* __restrict__ v = vs + (size_t)b * MROWS;
    const float* __restrict__ w = JR + (size_t)j * N_VERTS;
    float s0 = 0.f, s1 = 0.f, s2 = 0.f;
    for (int n = tid; n < N_VERTS; n += 256) {
        const float ww = w[n];
        s0 += ww * v[n * 3 + 0];
        s1 += ww * v[n * 3 + 1];
        s2 += ww * v[n * 3 + 2];
    }
    r0[tid] = s0; r1[tid] = s1; r2[tid] = s2;
    __syncthreads();
    for (int off = 128; off > 0; off >>= 1) {
        if (tid < off) {
            r0[tid] += r0[tid + off];
            r1[tid] += r1[tid + off];
            r2[tid] += r2[tid + off];
        }
        __syncthreads();
    }
    if (tid == 0) {
        float* o = Jout + ((size_t)b * N_JOINTS + j) * 3;
        o[0] = r0[0]; o[1] = r1[0]; o[2] = r2[0];
    }
}

// --- Kernel 3: per-batch rigging. One thread per batch element:
// quats -> R, pose_feat, kinematic chain G, G_final = G - G@rest, J_transformed.
__global__ void rig_kernel(const float* __restrict__ pose,
                           const float* __restrict__ betas,
                           const float* __restrict__ trans,
                           const float* __restrict__ J,
                           float* __restrict__ pf,     // (B,96), zero-padded tail
                           float* __restrict__ Graw,   // (B,24,12)
                           float* __restrict__ Gfin,   // (B,24,12)
                           float* __restrict__ Jt) {   // (B,24,3)
    const int b = blockIdx.x * blockDim.x + threadIdx.x;
    if (b >= BATCH) return;
    const int par[N_JOINTS] = {-1, 0, 0, 0, 1, 2, 3, 4, 5, 6, 7, 8,
                               9, 9, 9, 12, 13, 14, 16, 17, 18, 19, 20, 21};
    const float* th = pose + (size_t)b * 72;
    const float* Jb = J + (size_t)b * N_JOINTS * 3;
    float* Gr  = Graw + (size_t)b * N_JOINTS * 12;
    float* Gf  = Gfin + (size_t)b * N_JOINTS * 12;
    float* pfb = pf + (size_t)b * 96;
    const float t0 = trans[b * 3 + 0], t1 = trans[b * 3 + 1], t2 = trans[b * 3 + 2];

    for (int j = 0; j < N_JOINTS; ++j) {
        const float tx = th[3 * j], ty = th[3 * j + 1], tz = th[3 * j + 2];
        const float ax = tx + 1e-8f, ay = ty + 1e-8f, az = tz + 1e-8f;
        const float ang = sqrtf(ax * ax + ay * ay + az * az);
        const float inv = 1.0f / ang;
        const float half = ang * 0.5f;
        const float s = sinf(half), c = cosf(half);
        float x = s * tx * inv, y = s * ty * inv, z = s * tz * inv, w = c;
        // pose_feat uses UNnormalized (s*n, c-1); skip joint 0
        if (j >= 1) {
            pfb[4 * (j - 1) + 0] = x;
            pfb[4 * (j - 1) + 1] = y;
            pfb[4 * (j - 1) + 2] = z;
            pfb[4 * (j - 1) + 3] = c - 1.0f;
        }
        // normalize quat for the rotation matrix
        const float qn = 1.0f / sqrtf(w * w + x * x + y * y + z * z);
        w *= qn; x *= qn; y *= qn; z *= qn;
        const float w2 = w * w, x2 = x * x, y2 = y * y, z2 = z * z;
        const float wx = w * x, wy = w * y, wz = w * z;
        const float xy = x * y, xz = x * z, yz = y * z;
        float L[12];
        L[0] = w2 + x2 - y2 - z2; L[1] = 2.f * (xy - wz);     L[2]  = 2.f * (wy + xz);
        L[4] = 2.f * (wz + xy);   L[5] = w2 - x2 + y2 - z2;   L[6]  = 2.f * (yz - wx);
        L[8] = 2.f * (xz - wy);   L[9] = 2.f * (wx + yz);     L[10] = w2 - x2 - y2 + z2;
        const float jx = Jb[j * 3 + 0], jy = Jb[j * 3 + 1], jz = Jb[j * 3 + 2];
        if (par[j] < 0) {
            L[3] = jx; L[7] = jy; L[11] = jz;
        } else {
            L[3]  = jx - Jb[par[j] * 3 + 0];
            L[7]  = jy - Jb[par[j] * 3 + 1];
            L[11] = jz - Jb[par[j] * 3 + 2];
        }
        float G[12];
        if (par[j] < 0) {
#pragma unroll
            for (int r = 0; r < 12; ++r) G[r] = L[r];
        } else {
            float P[12];
#pragma unroll
            for (int r = 0; r < 12; ++r) P[r] = Gr[par[j] * 12 + r];  // same-thread RAW
#pragma unroll
            for (int r = 0; r < 3; ++r) {
#pragma unroll
                for (int cc = 0; cc < 4; ++cc) {
                    float acc = (cc == 3) ? P[r * 4 + 3] : 0.0f;
                    acc += P[r * 4 + 0] * L[0 * 4 + cc];
                    acc += P[r * 4 + 1] * L[1 * 4 + cc];
                    acc += P[r * 4 + 2] * L[2 * 4 + cc];
                    G[r * 4 + cc] = acc;
                }
            }
        }
#pragma unroll
        for (int r = 0; r < 12; ++r) Gr[j * 12 + r] = G[r];
        // J_transformed = chain translation + trans
        Jt[((size_t)b * N_JOINTS + j) * 3 + 0] = G[3]  + t0;
        Jt[((size_t)b * N_JOINTS + j) * 3 + 1] = G[7]  + t1;
        Jt[((size_t)b * N_JOINTS + j) * 3 + 2] = G[11] + t2;
        // G_final = G - G@rest : translation -= R_g @ J_j
#pragma unroll
        for (int r = 0; r < 3; ++r) {
            Gf[j * 12 + r * 4 + 0] = G[r * 4 + 0];
            Gf[j * 12 + r * 4 + 1] = G[r * 4 + 1];
            Gf[j * 12 + r * 4 + 2] = G[r * 4 + 2];
            Gf[j * 12 + r * 4 + 3] = G[r * 4 + 3] -
                (G[r * 4 + 0] * jx + G[r * 4 + 1] * jy + G[r * 4 + 2] * jz);
        }
    }
    pfb[92] = betas[(size_t)b * 300 + 1];
    pfb[93] = 0.f; pfb[94] = 0.f; pfb[95] = 0.f;   // zero-pad for WMMA K tail
}

// --- Kernel 4: v_posed[b, m] = v_shaped[b, m] + (posedirs @ pose_feat^T)[m, b]
__global__ void vposed_kernel(const float* __restrict__ PD,
                              const float* __restrict__ PF,
                              const float* __restrict__ vs,
                              float* __restrict__ out_vp,
                              int Mtiles, int totalTiles) {
    const int wid  = threadIdx.x >> 5;
    const int lane = threadIdx.x & 31;
    const int tile = blockIdx.x * (blockDim.x >> 5) + wid;
    if (tile >= totalTiles) return;
    const int mt = tile % Mtiles;
    const int nt = tile / Mtiles;
    const int m0 = mt * 16, n0 = nt * 32;

    v8f acc0, acc1;
    wmma_tile2_f32<KPOSE, KPOSE, 96>(PD, PF, MROWS, m0, n0, lane, acc0, acc1);

    const int col = lane & 15;
    const int hi  = lane >> 4;
    const int mb  = m0 + hi * 8;
    const size_t i0 = (size_t)(n0 + col) * MROWS + mb;
    const size_t i1 = (size_t)(n0 + 16 + col) * MROWS + mb;
    if (m0 + 16 <= MROWS) {
#pragma unroll
        for (int r = 0; r < 8; ++r) {
            out_vp[i0 + r] = acc0[r] + vs[i0 + r];
            out_vp[i1 + r] = acc1[r] + vs[i1 + r];
        }
    } else {
#pragma unroll
        for (int r = 0; r < 8; ++r) {
            if (mb + r < MROWS) {
                out_vp[i0 + r] = acc0[r] + vs[i0 + r];
                out_vp[i1 + r] = acc1[r] + vs[i1 + r];
            }
        }
    }
}

// --- Kernel 5: skinning. T = weights @ G_final, v = T * [v_posed,1] + trans
__global__ void skin_kernel(const float* __restrict__ Gfin,
                            const float* __restrict__ weights,
                            const float* __restrict__ vp,
                            const float* __restrict__ trans,
                            float* __restrict__ out_v) {
    __shared__ float sG[N_JOINTS * 12];
    const int b = blockIdx.y;
    for (int i = threadIdx.x; i < N_JOINTS * 12; i += 256)
        sG[i] = Gfin[(size_t)b * N_JOINTS * 12 + i];
    __syncthreads();
    const int v = blockIdx.x * 256 + threadIdx.x;
    if (v >= N_VERTS) return;
    float T[12];
#pragma unroll
    for (int r = 0; r < 12; ++r) T[r] = 0.f;
    const float* __restrict__ wv = weights + (size_t)v * N_JOINTS;
    for (int j = 0; j < N_JOINTS; ++j) {
        const float w = wv[j];
#pragma unroll
        for (int r = 0; r < 12; ++r) T[r] += w * sG[j * 12 + r];
    }
    const size_t base = (size_t)b * MROWS + (size_t)v * 3;
    const float px = vp[base + 0], py = vp[base + 1], pz = vp[base + 2];
    out_v[base + 0] = T[0] * px + T[1] * py + T[2]  * pz + T[3]  + trans[b * 3 + 0];
    out_v[base + 1] = T[4] * px + T[5] * py + T[6]  * pz + T[7]  + trans[b * 3 + 1];
    out_v[base + 2] = T[8] * px + T[9] * py + T[10] * pz + T[11] + trans[b * 3 + 2];
}

extern "C" void kernel_launch(void* const* d_in, const int* in_sizes, int n_in,
                              void* d_out, int out_size, void* d_ws, size_t ws_size,
                              hipStream_t stream) {
    const float* pose     = (const float*)d_in[0];
    const float* betas    = (const float*)d_in[1];
    const float* trans    = (const float*)d_in[2];
    const float* v_templ  = (const float*)d_in[3];
    const float* shapedir = (const float*)d_in[4];
    const float* posedirs = (const float*)d_in[5];
    const float* J_reg    = (const float*)d_in[6];
    const float* weights  = (const float*)d_in[7];

    float* out     = (float*)d_out;
    float* out_v   = out;                                 // (B, N, 3)
    float* out_vp  = out + (size_t)BATCH * MROWS;         // (B, N, 3)
    float* out_vs  = out + (size_t)2 * BATCH * MROWS;     // (B, N, 3)
    float* out_Jt  = out + (size_t)3 * BATCH * MROWS;     // (B, 24, 3)

    float* ws   = (float*)d_ws;
    float* Jbuf = ws;                                     // B*24*3   = 36864
    float* pf   = Jbuf + BATCH * N_JOINTS * 3;            // B*96     = 49152
    float* Graw = pf + BATCH * 96;                        // B*24*12  = 147456
    float* Gfin = Graw + BATCH * N_JOINTS * 12;           // B*24*12  = 147456

    const int Mtiles = (MROWS + 15) / 16;                 // 1292
    const int Ntiles = BATCH / 32;                        // 16 (16x32 tiles)
    const int totalTiles = Mtiles * Ntiles;               // 20672
    const int wavesPerBlock = 8;                          // 256 threads, wave32
    const int gemmGrid = (totalTiles + wavesPerBlock - 1) / wavesPerBlock;

    vshaped_kernel<<<gemmGrid, 256, 0, stream>>>(shapedir, betas, v_templ,
                                                 out_vs, Mtiles, totalTiles);
    jreg_kernel<<<BATCH * N_JOINTS, 256, 0, stream>>>(out_vs, J_reg, Jbuf);
    rig_kernel<<<(BATCH + 255) / 256, 256, 0, stream>>>(pose, betas, trans, Jbuf,
                                                        pf, Graw, Gfin, out_Jt);
    vposed_kernel<<<gemmGrid, 256, 0, stream>>>(posedirs, pf, out_vs,
                                                out_vp, Mtiles, totalTiles);
    dim3 sg((N_VERTS + 255) / 256, BATCH);
    skin_kernel<<<sg, 256, 0, stream>>>(Gfin, weights, out_vp, trans, out_v);
}